// MeshInterpolator_18554258719137
// MI455X (gfx1250) — compile-verified
//
#include <hip/hip_runtime.h>
#include <stdint.h>

#define NMESH 128
#define NCH 16
#define SPACING_F 0.1f
#define CH_STRIDE (NMESH * NMESH * NMESH) /* 2097152 floats = 8 MB per channel */

typedef uint32_t u32x4 __attribute__((ext_vector_type(4)));
typedef int      i32x4 __attribute__((ext_vector_type(4)));
typedef int      i32x8 __attribute__((ext_vector_type(8)));
typedef float    f32x4 __attribute__((ext_vector_type(4)));

// One thread = one point. Each wave (32 points) stages its 96 point-floats
// into LDS via the Tensor Data Mover (TENSORcnt-tracked DMA), then each lane
// does the 27-neighbor x 16-channel gather-FMA from the L2-resident mesh.
// Channel bases are pinned in SGPR pairs so each tap's byte offset VGPR is
// shared by all 16 channel loads (saddr form), minimizing VALU issue.
__global__ __launch_bounds__(256) void mesh_interp_tsc_kernel(
    const float* __restrict__ mesh,    // [16][128][128][128]
    const float* __restrict__ points,  // [N][3]
    float* __restrict__ out,           // [N][16]
    int npts)
{
    __shared__ float spts[256 * 3];

    const int tid  = (int)threadIdx.x;
    const int lane = tid & 31;
    const int wave = tid >> 5;
    const int n0   = (int)blockIdx.x * 256 + wave * 32;  // wave-uniform

    // ---- TDM: DMA this wave's 32 points (96 f32) from global into LDS ----
    {
        const int elem0 = n0 * 3;                  // fits int32 (<= 300k)
        int rem = npts * 3 - elem0;                // OOB bound for last wave
        if (rem < 0) rem = 0;
        const uint32_t d0 = (uint32_t)rem;         // tensor_dim0: OOB reads -> 0

        const uint64_t gaddr   = (uint64_t)(uintptr_t)points + (uint32_t)(elem0 * 4);
        const uint32_t ldsaddr = (uint32_t)(uintptr_t)(&spts[wave * 96]);

        // D# group 0 (128b): count=1 | lds_addr | global_addr[56:0] | type=2
        u32x4 g0;
        g0[0] = 1u;
        g0[1] = ldsaddr;
        g0[2] = (uint32_t)(gaddr & 0xFFFFFFFFu);
        g0[3] = (uint32_t)((gaddr >> 32) & 0x01FFFFFFu) | (2u << 30);

        // D# group 1 (256b): data_size=4B, tensor_dim0=rem, tensor_dim1=1,
        // tile_dim0=96, tile_dim1=1, benign strides, no multicast/pad/iterate.
        i32x8 g1;
        g1[0] = (int)(2u << 16);                    // data_size = 2 -> 4 bytes
        g1[1] = (int)((d0 & 0xFFFFu) << 16);        // tensor_dim0[15:0]
        g1[2] = (int)((d0 >> 16) | (1u << 16));     // tensor_dim0[31:16] | tensor_dim1=1
        g1[3] = (int)(96u << 16);                   // tile_dim0 = 96
        g1[4] = 1;                                  // tile_dim1 = 1, tile_dim2 = 0
        g1[5] = (int)d0;                            // tensor_dim0_stride lo (1D: unused)
        g1[6] = 0;
        g1[7] = 0;

        i32x4 z4 = {0, 0, 0, 0};
#if __clang_major__ >= 23
        i32x8 z8 = {0, 0, 0, 0, 0, 0, 0, 0};
        __builtin_amdgcn_tensor_load_to_lds(g0, g1, z4, z4, z8, 0);
#else
        __builtin_amdgcn_tensor_load_to_lds(g0, g1, z4, z4, 0);
#endif
        __builtin_amdgcn_s_wait_tensorcnt(0);
        __asm__ volatile("" ::: "memory");  // keep LDS reads below the wait
    }

    const int n = n0 + lane;
    if (n >= npts) return;

    // Pin 16 per-channel base pointers in SGPR pairs (opaque to the folder):
    // forces global_load saddr form with one shared voffset per stencil tap.
    const float* chan[NCH];
#pragma unroll
    for (int c = 0; c < NCH; ++c) {
        const float* p = mesh + (size_t)c * CH_STRIDE;
        __asm__("" : "+s"(p));
        chan[c] = p;
    }

    const float px = spts[wave * 96 + lane * 3 + 0];
    const float py = spts[wave * 96 + lane * 3 + 1];
    const float pz = spts[wave * 96 + lane * 3 + 2];

    // pc = p / SPACING ; rp = round-to-nearest-even ; dist = pc - rp
    const float pcx = px / SPACING_F, pcy = py / SPACING_F, pcz = pz / SPACING_F;
    const float rx = rintf(pcx), ry = rintf(pcy), rz = rintf(pcz);
    const float dx = pcx - rx, dy = pcy - ry, dz = pcz - rz;
    const int ix = (int)rx, iy = (int)ry, iz = (int)rz;

    // TSC weights: (2d-1)^2/8, 0.75-d^2, (2d+1)^2/8
    const float wx[3] = { (2.f*dx - 1.f)*(2.f*dx - 1.f)*0.125f, 0.75f - dx*dx,
                          (2.f*dx + 1.f)*(2.f*dx + 1.f)*0.125f };
    const float wy[3] = { (2.f*dy - 1.f)*(2.f*dy - 1.f)*0.125f, 0.75f - dy*dy,
                          (2.f*dy + 1.f)*(2.f*dy + 1.f)*0.125f };
    const float wz[3] = { (2.f*dz - 1.f)*(2.f*dz - 1.f)*0.125f, 0.75f - dz*dz,
                          (2.f*dz + 1.f)*(2.f*dz + 1.f)*0.125f };

    // (i +/- 1) & 127 == (i +/- 1) mod 128 for the i in [0,128] that occur
    const uint32_t xi[3] = { (uint32_t)((ix - 1) & (NMESH - 1)),
                             (uint32_t)(ix & (NMESH - 1)),
                             (uint32_t)((ix + 1) & (NMESH - 1)) };
    const uint32_t yi[3] = { (uint32_t)((iy - 1) & (NMESH - 1)),
                             (uint32_t)(iy & (NMESH - 1)),
                             (uint32_t)((iy + 1) & (NMESH - 1)) };
    const uint32_t zi[3] = { (uint32_t)((iz - 1) & (NMESH - 1)),
                             (uint32_t)(iz & (NMESH - 1)),
                             (uint32_t)((iz + 1) & (NMESH - 1)) };

    float acc[NCH];
#pragma unroll
    for (int c = 0; c < NCH; ++c) acc[c] = 0.f;

#pragma unroll
    for (int a = 0; a < 3; ++a) {
#pragma unroll
        for (int b = 0; b < 3; ++b) {
            const float    wab  = wx[a] * wy[b];
            const uint32_t base = (xi[a] * NMESH + yi[b]) * NMESH;
#pragma unroll
            for (int cz = 0; cz < 3; ++cz) {
                const float    w   = wab * wz[cz];
                const uint32_t lin = base + zi[cz];  // shared divergent voffset
#pragma unroll
                for (int c = 0; c < NCH; ++c)
                    acc[c] = fmaf(w, chan[c][lin], acc[c]);
            }
        }
    }

    // 64 contiguous bytes per point -> 4x b128 non-temporal stores:
    // streamed output must not evict the L2-resident mesh (134 MB of 192 MB).
    f32x4* o4 = reinterpret_cast<f32x4*>(out + (size_t)n * NCH);
    f32x4 v0 = { acc[0],  acc[1],  acc[2],  acc[3]  };
    f32x4 v1 = { acc[4],  acc[5],  acc[6],  acc[7]  };
    f32x4 v2 = { acc[8],  acc[9],  acc[10], acc[11] };
    f32x4 v3 = { acc[12], acc[13], acc[14], acc[15] };
    __builtin_nontemporal_store(v0, o4 + 0);
    __builtin_nontemporal_store(v1, o4 + 1);
    __builtin_nontemporal_store(v2, o4 + 2);
    __builtin_nontemporal_store(v3, o4 + 3);
}

extern "C" void kernel_launch(void* const* d_in, const int* in_sizes, int n_in,
                              void* d_out, int out_size, void* d_ws, size_t ws_size,
                              hipStream_t stream) {
    const float* mesh   = (const float*)d_in[0];  // 16*128^3 floats
    const float* points = (const float*)d_in[1];  // N*3 floats
    float* out          = (float*)d_out;          // N*16 floats

    const int npts   = in_sizes[1] / 3;
    const int block  = 256;
    const int nblock = (npts + block - 1) / block;

    mesh_interp_tsc_kernel<<<nblock, block, 0, stream>>>(mesh, points, out, npts);
}